// IDKN_cat_7722351198325
// MI455X (gfx1250) — compile-verified
//
#include <hip/hip_runtime.h>
#include <hip/hip_bf16.h>
#include <math.h>

// ---------------------------------------------------------------------------
// MI455X reasoning: edge scatter/gather dominates (~1 GB of L2-resident
// traffic); dense transforms are ~110 MFLOP total. So edge kernels are plain
// coalesced f32-atomic passes (resolved in L2), and the node transforms use
// the fp32 WMMA path (v_wmma_f32_16x16x4_f32, wave32, 16-node tiles) which
// keeps fp32 precision and uses the matrix pipe for the dense part.
// ---------------------------------------------------------------------------

typedef __attribute__((ext_vector_type(2))) float v2f;
typedef __attribute__((ext_vector_type(8))) float v8f;

#define LEAKY_SLOPE 0.2f

__device__ __forceinline__ float leaky_relu(float x) {
    return x > 0.0f ? x : LEAKY_SLOPE * x;
}

// float atomic max via monotonic int/uint bit trick (init must be -inf)
__device__ __forceinline__ void atomic_max_f32(float* addr, float v) {
    if (v >= 0.0f) atomicMax((int*)addr, __float_as_int(v));
    else           atomicMin((unsigned int*)addr, __float_as_uint(v));
}

// extended edge list: e in [0,E) = real edges, e in [E, E+N) = self loops
__device__ __forceinline__ void edge_ends(const int* __restrict__ ei, int E,
                                          int e, int& s, int& d) {
    if (e < E) { s = ei[e]; d = ei[E + e]; }
    else       { s = e - E; d = s; }
}

// ---------------------------------------------------------------------------
// lin1: out[N,8] = concat(x1[N,7], x2[N,10]) @ w[17,8] + b   (K padded to 20)
// One wave32 per 16-node tile; V_WMMA_F32_16X16X4_F32 A layout:
//   lanes 0-15: M=lane, K = kb+{0,1};  lanes 16-31: M=lane-16, K = kb+{2,3}
// C/D layout: VGPR r -> (M = r + 8*(lane>=16), N = lane&15)
// ---------------------------------------------------------------------------
__global__ void lin1_wmma(const float* __restrict__ x1, const float* __restrict__ x2,
                          const float* __restrict__ w, const float* __restrict__ b,
                          float* __restrict__ out, int n) {
    int wave  = (blockIdx.x * blockDim.x + threadIdx.x) >> 5;
    int lane  = threadIdx.x & 31;
    int row0  = wave * 16;
    if (row0 >= n) return;
    int m     = lane & 15;
    int khalf = lane >> 4;
    int node  = row0 + m; if (node >= n) node = n - 1;

    v8f acc = {};
#pragma unroll
    for (int kb = 0; kb < 20; kb += 4) {
        int k0 = kb + 2 * khalf;
        int k1 = k0 + 1;
        v2f a, bb;
        a.x = (k0 < 7) ? x1[node * 7 + k0] : (k0 < 17 ? x2[node * 10 + (k0 - 7)] : 0.0f);
        a.y = (k1 < 7) ? x1[node * 7 + k1] : (k1 < 17 ? x2[node * 10 + (k1 - 7)] : 0.0f);
        bb.x = (k0 < 17 && m < 8) ? w[k0 * 8 + m] : 0.0f;
        bb.y = (k1 < 17 && m < 8) ? w[k1 * 8 + m] : 0.0f;
        acc = __builtin_amdgcn_wmma_f32_16x16x4_f32(false, a, false, bb,
                                                    (short)0, acc, false, false);
    }
    if (m < 8) {
#pragma unroll
        for (int r = 0; r < 8; ++r) {
            int nd = row0 + r + 8 * khalf;
            if (nd < n) out[nd * 8 + m] = acc[r] + b[m];
        }
    }
}

// ---------------------------------------------------------------------------
// GAT node transform: xl[N,KOUT] = x[N,KIN] @ w[KIN,KOUT]    (KIN % 4 == 0)
// ---------------------------------------------------------------------------
template <int KIN, int KOUT>
__global__ void gat_transform_wmma(const float* __restrict__ x, const float* __restrict__ w,
                                   float* __restrict__ xl, int n) {
    int wave  = (blockIdx.x * blockDim.x + threadIdx.x) >> 5;
    int lane  = threadIdx.x & 31;
    int row0  = wave * 16;
    if (row0 >= n) return;
    int m     = lane & 15;
    int khalf = lane >> 4;
    int node  = row0 + m; if (node >= n) node = n - 1;

    constexpr int NB = (KOUT + 15) / 16;
#pragma unroll
    for (int nb = 0; nb < NB; ++nb) {
        int col = nb * 16 + m;
        v8f acc = {};
#pragma unroll
        for (int kb = 0; kb < KIN; kb += 4) {
            int k0 = kb + 2 * khalf;
            v2f a, bb;
            a.x = x[node * KIN + k0];
            a.y = x[node * KIN + k0 + 1];
            bb.x = (col < KOUT) ? w[k0 * KOUT + col] : 0.0f;
            bb.y = (col < KOUT) ? w[(k0 + 1) * KOUT + col] : 0.0f;
            acc = __builtin_amdgcn_wmma_f32_16x16x4_f32(false, a, false, bb,
                                                        (short)0, acc, false, false);
        }
        if (col < KOUT) {
#pragma unroll
            for (int r = 0; r < 8; ++r) {
                int nd = row0 + r + 8 * khalf;
                if (nd < n) xl[nd * KOUT + col] = acc[r];
            }
        }
    }
}

// per-node attention logits: als[i,h] = xl[i,h,:]·a_src[h,:], ald likewise
template <int H, int C>
__global__ void node_attn(const float* __restrict__ xl, const float* __restrict__ asrc,
                          const float* __restrict__ adst, float* __restrict__ als,
                          float* __restrict__ ald, int n) {
    int i = blockIdx.x * blockDim.x + threadIdx.x;
    if (i >= n) return;
#pragma unroll
    for (int h = 0; h < H; ++h) {
        float s = 0.0f, d = 0.0f;
#pragma unroll
        for (int c = 0; c < C; ++c) {
            float v = xl[i * H * C + h * C + c];
            s += v * asrc[h * C + c];
            d += v * adst[h * C + c];
        }
        als[i * H + h] = s;
        ald[i * H + h] = d;
    }
}

template <int H, int HC>
__global__ void init_layer(float* __restrict__ m, float* __restrict__ den,
                           float* __restrict__ agg, int n) {
    int i = blockIdx.x * blockDim.x + threadIdx.x;
    if (i >= n) return;
#pragma unroll
    for (int h = 0; h < H; ++h) {
        m[i * H + h]   = -INFINITY;
        den[i * H + h] = 0.0f;
    }
#pragma unroll
    for (int r = 0; r < HC; ++r) agg[i * HC + r] = 0.0f;
}

template <int H>
__global__ void edge_max(const int* __restrict__ ei, int E, int Etot,
                         const float* __restrict__ als, const float* __restrict__ ald,
                         float* __restrict__ m) {
    int e = blockIdx.x * blockDim.x + threadIdx.x;
    if (e >= Etot) return;
    int s, d; edge_ends(ei, E, e, s, d);
#pragma unroll
    for (int h = 0; h < H; ++h) {
        float eh = leaky_relu(als[s * H + h] + ald[d * H + h]);
        atomic_max_f32(&m[d * H + h], eh);
    }
}

template <int H>
__global__ void edge_den(const int* __restrict__ ei, int E, int Etot,
                         const float* __restrict__ als, const float* __restrict__ ald,
                         const float* __restrict__ m, float* __restrict__ den) {
    int e = blockIdx.x * blockDim.x + threadIdx.x;
    if (e >= Etot) return;
    int s, d; edge_ends(ei, E, e, s, d);
#pragma unroll
    for (int h = 0; h < H; ++h) {
        float eh = leaky_relu(als[s * H + h] + ald[d * H + h]);
        atomicAdd(&den[d * H + h], __expf(eh - m[d * H + h]));
    }
}

// H*C lanes per edge: lane r handles feature r (head h = r/C). Coalesced
// gather of xl[src] and coalesced f32 atomics into agg[dst] (resolved in L2).
template <int H, int C>
__global__ void edge_agg(const int* __restrict__ ei, int E, int Etot,
                         const float* __restrict__ als, const float* __restrict__ ald,
                         const float* __restrict__ m, const float* __restrict__ den,
                         const float* __restrict__ xl, float* __restrict__ agg) {
    int t = blockIdx.x * blockDim.x + threadIdx.x;
    int e = t / (H * C);
    int r = t % (H * C);
    if (e >= Etot) return;
    int h = r / C;
    int s, d; edge_ends(ei, E, e, s, d);
    float eh    = leaky_relu(als[s * H + h] + ald[d * H + h]);
    float ex    = __expf(eh - m[d * H + h]);
    float alpha = ex / (den[d * H + h] + 1e-16f);
    atomicAdd(&agg[d * H * C + r], alpha * xl[s * H * C + r]);
}

template <int HC>
__global__ void add_bias(float* __restrict__ x, const float* __restrict__ b, int n) {
    int t = blockIdx.x * blockDim.x + threadIdx.x;
    if (t >= n * HC) return;
    x[t] += b[t % HC];
}

__global__ void init_final(float* __restrict__ deg, float* __restrict__ local,
                           float* __restrict__ r3, int n) {
    int i = blockIdx.x * blockDim.x + threadIdx.x;
    if (i >= n) return;
    deg[i] = 0.0f;
    local[i] = 0.0f;
#pragma unroll
    for (int c = 0; c < 8; ++c) r3[i * 8 + c] = 0.0f;
}

__global__ void final_edge(const int* __restrict__ ei, int E, int Etot,
                           const float* __restrict__ x6, float* __restrict__ deg,
                           float* __restrict__ local, float* __restrict__ r3) {
    int e = blockIdx.x * blockDim.x + threadIdx.x;
    if (e >= Etot) return;
    int s, d; edge_ends(ei, E, e, s, d);
    float dot = 0.0f;
    float xd[8];
#pragma unroll
    for (int c = 0; c < 8; ++c) {
        xd[c] = x6[d * 8 + c];
        dot += x6[s * 8 + c] * xd[c];
    }
    atomicAdd(&local[s], dot);
#pragma unroll
    for (int c = 0; c < 8; ++c) atomicAdd(&r3[s * 8 + c], xd[c]);
    atomicAdd(&deg[s], 1.0f);
}

__global__ void final_out(const float* __restrict__ local, const float* __restrict__ r3,
                          const float* __restrict__ deg, const float* __restrict__ lin2w,
                          float* __restrict__ out, int n) {
    int i = blockIdx.x * blockDim.x + threadIdx.x;
    if (i >= n) return;
    float g = 0.0f;
#pragma unroll
    for (int c = 0; c < 8; ++c) g += r3[i * 8 + c] * lin2w[c];
    // local/deg + (r3/deg)@w == (local_sum + r3_sum@w) / deg
    out[i] = (local[i] + g) / deg[i];
}

// ---------------------------------------------------------------------------

static inline int cdiv(long a, long b) { return (int)((a + b - 1) / b); }

extern "C" void kernel_launch(void* const* d_in, const int* in_sizes, int n_in,
                              void* d_out, int out_size, void* d_ws, size_t ws_size,
                              hipStream_t stream) {
    const float* x1     = (const float*)d_in[0];
    const float* x2     = (const float*)d_in[1];
    const float* lin1w  = (const float*)d_in[2];
    const float* lin1b  = (const float*)d_in[3];
    const float* W1     = (const float*)d_in[4];
    const float* as1    = (const float*)d_in[5];
    const float* ad1    = (const float*)d_in[6];
    const float* b1     = (const float*)d_in[7];
    const float* W2     = (const float*)d_in[8];
    const float* as2    = (const float*)d_in[9];
    const float* ad2    = (const float*)d_in[10];
    const float* b2     = (const float*)d_in[11];
    const float* W3     = (const float*)d_in[12];
    const float* as3    = (const float*)d_in[13];
    const float* ad3    = (const float*)d_in[14];
    const float* b3     = (const float*)d_in[15];
    const float* lin2w  = (const float*)d_in[16];
    const int*   ei     = (const int*)d_in[17];

    const int n    = in_sizes[0] / 7;    // 50000
    const int E    = in_sizes[17] / 2;   // 1600000
    const int Etot = E + n;              // + self loops

    // workspace layout (floats): A,B ping-pong node features; attn buffers
    float* ws    = (float*)d_ws;
    float* A     = ws;                    // N*32 (current features / agg target)
    float* B     = A    + (size_t)n * 32; // N*32 (xl)
    float* als   = B    + (size_t)n * 32; // N*8
    float* ald   = als  + (size_t)n * 8;  // N*8
    float* mbuf  = ald  + (size_t)n * 8;  // N*8
    float* den   = mbuf + (size_t)n * 8;  // N*8
    float* deg   = den  + (size_t)n * 8;  // N
    float* local = deg  + n;              // N
    float* r3    = local + n;             // N*8

    const int T = 256;
    const int gN  = cdiv(n, T);
    const int gE  = cdiv(Etot, T);
    const int gW  = cdiv((long)cdiv(n, 16) * 32, T);  // wave-per-16-node-tile

    // lin1: concat + [17x8] GEMM (WMMA f32) -> A[N,8]
    lin1_wmma<<<gW, T, 0, stream>>>(x1, x2, lin1w, lin1b, A, n);

    // ---- GAT layer 1: KIN=8, KOUT=32, H=8, C=4 ----
    gat_transform_wmma<8, 32><<<gW, T, 0, stream>>>(A, W1, B, n);
    node_attn<8, 4><<<gN, T, 0, stream>>>(B, as1, ad1, als, ald, n);
    init_layer<8, 32><<<gN, T, 0, stream>>>(mbuf, den, A, n);
    edge_max<8><<<gE, T, 0, stream>>>(ei, E, Etot, als, ald, mbuf);
    edge_den<8><<<gE, T, 0, stream>>>(ei, E, Etot, als, ald, mbuf, den);
    edge_agg<8, 4><<<cdiv((long)Etot * 32, T), T, 0, stream>>>(ei, E, Etot, als, ald,
                                                               mbuf, den, B, A);
    add_bias<32><<<cdiv((long)n * 32, T), T, 0, stream>>>(A, b1, n);

    // ---- GAT layer 2: KIN=32, KOUT=16, H=4, C=4 ----
    gat_transform_wmma<32, 16><<<gW, T, 0, stream>>>(A, W2, B, n);
    node_attn<4, 4><<<gN, T, 0, stream>>>(B, as2, ad2, als, ald, n);
    init_layer<4, 16><<<gN, T, 0, stream>>>(mbuf, den, A, n);
    edge_max<4><<<gE, T, 0, stream>>>(ei, E, Etot, als, ald, mbuf);
    edge_den<4><<<gE, T, 0, stream>>>(ei, E, Etot, als, ald, mbuf, den);
    edge_agg<4, 4><<<cdiv((long)Etot * 16, T), T, 0, stream>>>(ei, E, Etot, als, ald,
                                                               mbuf, den, B, A);
    add_bias<16><<<cdiv((long)n * 16, T), T, 0, stream>>>(A, b2, n);

    // ---- GAT layer 3: KIN=16, KOUT=8, H=2, C=4 ----
    gat_transform_wmma<16, 8><<<gW, T, 0, stream>>>(A, W3, B, n);
    node_attn<2, 4><<<gN, T, 0, stream>>>(B, as3, ad3, als, ald, n);
    init_layer<2, 8><<<gN, T, 0, stream>>>(mbuf, den, A, n);
    edge_max<2><<<gE, T, 0, stream>>>(ei, E, Etot, als, ald, mbuf);
    edge_den<2><<<gE, T, 0, stream>>>(ei, E, Etot, als, ald, mbuf, den);
    edge_agg<2, 4><<<cdiv((long)Etot * 8, T), T, 0, stream>>>(ei, E, Etot, als, ald,
                                                              mbuf, den, B, A);
    add_bias<8><<<cdiv((long)n * 8, T), T, 0, stream>>>(A, b3, n);

    // ---- final scoring ----
    init_final<<<gN, T, 0, stream>>>(deg, local, r3, n);
    final_edge<<<gE, T, 0, stream>>>(ei, E, Etot, A, deg, local, r3);
    final_out<<<gN, T, 0, stream>>>(local, r3, deg, lin2w, (float*)d_out, n);
}